// SimDense_58901181497981
// MI455X (gfx1250) — compile-verified
//
#include <hip/hip_runtime.h>
#include <stdint.h>

#define NPT      16          // nodes per thread (K = 16384 = 1024*16)
#define NTHREADS 1024        // 32 wave32 waves on one WGP
#define CCFL_C   0.9f
#define DX_C     0.01f

typedef unsigned int uint32x4 __attribute__((ext_vector_type(4)));
typedef int          int32x8  __attribute__((ext_vector_type(8)));
typedef int          int32x4  __attribute__((ext_vector_type(4)));

#ifndef __has_builtin
#define __has_builtin(x) 0
#endif

#if __has_builtin(__builtin_amdgcn_tensor_store_from_lds)
#define HAVE_TDM 1
#else
#define HAVE_TDM 0
#endif

#if HAVE_TDM
// DMA one contiguous row of `nelem` f32 from LDS (byte addr lds_addr) to gptr.
// D# per CDNA5 ISA ch.8: group0 {count, lds_addr, global_addr, type=2},
// group1 {data_size=4B, tensor_dim0=tile_dim0=nelem, tensor_dim1=tile_dim1=1, stride0=nelem}.
__device__ __forceinline__ void tdm_store_row(unsigned lds_addr, const float* gptr, int nelem) {
  unsigned long long ga = (unsigned long long)(uintptr_t)gptr;
  uint32x4 g0;
  g0[0] = 1u;                                                 // count=1, user mode
  g0[1] = lds_addr;                                           // LDS byte address
  g0[2] = (unsigned)(ga & 0xFFFFFFFFu);                       // global addr lo
  g0[3] = (unsigned)((ga >> 32) & 0x1FFFFFFu) | (2u << 30);   // addr hi | type=2
  unsigned n = (unsigned)nelem;
  int32x8 g1;
  g1[0] = (int)(2u << 16);                                    // data_size = 4 bytes
  g1[1] = (int)((n & 0xFFFFu) << 16);                         // tensor_dim0[15:0]
  g1[2] = (int)(((n >> 16) & 0xFFFFu) | (1u << 16));          // tensor_dim0[31:16] | tensor_dim1 lo = 1
  g1[3] = (int)((n & 0xFFFFu) << 16);                         // tensor_dim1 hi = 0 | tile_dim0 = n
  g1[4] = 1;                                                  // tile_dim1 = 1, tile_dim2 = 0
  g1[5] = (int)n;                                             // tensor_dim0_stride lo32
  g1[6] = 0;                                                  // stride0 hi | stride1 lo
  g1[7] = 0;                                                  // stride1 hi
  int32x4 z4 = {0, 0, 0, 0};                                  // groups 2/3 unused (<=2D)
  int32x8 z8 = {0, 0, 0, 0, 0, 0, 0, 0};
  // 6-arg form (clang-23 / therock-10.0): (g0, g1, g2, g3, extra, cpol)
  __builtin_amdgcn_tensor_store_from_lds(g0, g1, z4, z4, z8, 0);
}
#endif

__global__ __launch_bounds__(NTHREADS)
void sim_blood1d_kernel(const float* __restrict__ R1_param,
                        const float* __restrict__ A_init,
                        const float* __restrict__ Q_init,
                        const float* __restrict__ beta,
                        const float* __restrict__ A0,
                        const float* __restrict__ input_data,
                        const int*   __restrict__ strides,
                        float* __restrict__ out,
                        int K, int T)
{
  extern __shared__ float smem[];
  float* As  = smem;            // [K] area
  float* Qs  = As + K;          // [K] flow
  float* Pb  = Qs + K;          // [K] pressure tile (TDM source)
  float* inp = Pb + K;          // [T] inlet waveform
  __shared__ float red[32];
  __shared__ float s_dt;

  const int tid  = threadIdx.x;
  const int base = tid * NPT;
  const int lane = tid & 31;
  const int wid  = tid >> 5;

  // stage inlet waveform into LDS once
  for (int i = tid; i < T; i += NTHREADS) inp[i] = input_data[i];

  // per-node constants held in registers (beta, 1/A0); state -> LDS
  float bta[NPT], ia0[NPT];
#pragma unroll
  for (int j = 0; j < NPT; ++j) {
    int g = base + j;
    bta[j] = beta[g];
    ia0[j] = 1.0f / A0[g];
    float a = A_init[g];
    As[g] = a;
    Qs[g] = Q_init[g];
    Pb[g] = bta[j] * (sqrtf(a * ia0[j]) - 1.0f);   // pressure(A_init) for first q_out
  }
  // halo constants (edge-padded like beta_p / sqrtA0_p in the reference)
  int gL = (base == 0) ? 0 : base - 1;
  int gR = (base + NPT >= K) ? K - 1 : base + NPT;
  float btaL = beta[gL], ia0L = 1.0f / A0[gL];
  float btaR = beta[gR], ia0R = 1.0f / A0[gR];

  // r = softplus(R1); outlet resistance at node K-1
  float x  = R1_param[0];
  float r  = (x > 20.0f) ? x : log1pf(__expf(x));
  int  s0  = strides[2] - 2, s1 = strides[3] + 2;
  float invR1_last = 1.0f / ((((K - 1) >= s0) && ((K - 1) < s1)) ? r : 1.0f);

  const unsigned pbAddr = (unsigned)(uintptr_t)(void*)Pb;  // low 32 bits = LDS byte offset
  const float third = 1.0f / 3.0f;

  __syncthreads();

  for (int t = 0; t < T; ++t) {
    // ---- adaptive CFL: max(|u| + c) over all nodes ----
    float m = 0.0f;
#pragma unroll
    for (int j = 0; j < NPT; ++j) {
      int g = base + j;
      float a = As[g], q = Qs[g];
      float u = q / a;
      float c = sqrtf(bta[j] * sqrtf(a * ia0[j]) * 0.5f);   // rho = 1
      m = fmaxf(m, fabsf(u) + c);
    }
#pragma unroll
    for (int s = 16; s > 0; s >>= 1) m = fmaxf(m, __shfl_xor(m, s, 32));
    if (lane == 0) red[wid] = m;
    __syncthreads();
    if (wid == 0) {
      float v = red[lane];
#pragma unroll
      for (int s = 16; s > 0; s >>= 1) v = fmaxf(v, __shfl_xor(v, s, 32));
      if (lane == 0) s_dt = CCFL_C * DX_C / v;
    }
    __syncthreads();
    const float lam = s_dt * (0.5f / DX_C);

    const float q_in  = inp[t];
    const float q_out = Pb[K - 1] * invR1_last;   // P(current A)[K-1] / R1

    // ---- gather stencil with ghost states ----
    float av[NPT + 2], qv[NPT + 2];
    av[0] = As[gL];                                    // base==0 -> As[0] (ghost)
    qv[0] = (base == 0) ? q_in : Qs[base - 1];
#pragma unroll
    for (int j = 0; j < NPT; ++j) { av[j + 1] = As[base + j]; qv[j + 1] = Qs[base + j]; }
    av[NPT + 1] = As[gR];                              // base+NPT==K -> As[K-1] (ghost)
    qv[NPT + 1] = (base + NPT >= K) ? q_out : Qs[base + NPT];

    float f2v[NPT + 2];
#pragma unroll
    for (int xx = 0; xx < NPT + 2; ++xx) {
      float b  = (xx == 0) ? btaL : (xx == NPT + 1) ? btaR : bta[xx - 1];
      float ii = (xx == 0) ? ia0L : (xx == NPT + 1) ? ia0R : ia0[xx - 1];
      float a = av[xx], q = qv[xx];
      // F2 = Q^2/A + beta/(3 rho sqrt(A0)) * A^1.5 ; 1/sqrt(A0) = sqrt(1/A0)
      f2v[xx] = q * q / a + b * third * sqrtf(ii) * (a * sqrtf(a));
    }

    float an[NPT], qn[NPT];
#pragma unroll
    for (int j = 0; j < NPT; ++j) {
      float A2 = 0.5f * (av[j] + av[j + 2]) - lam * (qv[j + 2] - qv[j]);
      float Q2 = 0.5f * (qv[j] + qv[j + 2]) - lam * (f2v[j + 2] - f2v[j]);
      int g = base + j;
      if (g == 0)     Q2 = q_in;
      if (g == K - 1) Q2 = q_out;
      an[j] = fmaxf(A2, 1e-3f);
      qn[j] = Q2;
    }

    // previous step's TDM of Pb must drain before we overwrite the tile
#if HAVE_TDM
    if (wid == 0) __builtin_amdgcn_s_wait_tensorcnt(0);
#endif
    __syncthreads();

#pragma unroll
    for (int j = 0; j < NPT; ++j) {
      int g = base + j;
      As[g] = an[j];
      Qs[g] = qn[j];
      float pn = bta[j] * (sqrtf(an[j] * ia0[j]) - 1.0f);
      Pb[g] = pn;
#if !HAVE_TDM
      out[(size_t)t * K + g] = pn;     // fallback path: direct stores
#endif
    }
    __syncthreads();

#if HAVE_TDM
    if (wid == 0) tdm_store_row(pbAddr, out + (size_t)t * K, K);  // async DMA, overlaps next step
#endif
  }
#if HAVE_TDM
  if (wid == 0) __builtin_amdgcn_s_wait_tensorcnt(0);
#endif
}

extern "C" void kernel_launch(void* const* d_in, const int* in_sizes, int n_in,
                              void* d_out, int out_size, void* d_ws, size_t ws_size,
                              hipStream_t stream) {
  const float* R1_param   = (const float*)d_in[0];
  const float* A_init     = (const float*)d_in[1];
  const float* Q_init     = (const float*)d_in[2];
  const float* beta       = (const float*)d_in[3];
  const float* A0         = (const float*)d_in[4];
  const float* input_data = (const float*)d_in[5];
  const int*   strides    = (const int*)d_in[6];
  int K = in_sizes[1];
  int T = in_sizes[5];

  size_t shmem = (size_t)(3 * (size_t)K + (size_t)T) * sizeof(float);  // ~198 KB < 320 KB WGP LDS
  (void)hipFuncSetAttribute(reinterpret_cast<const void*>(sim_blood1d_kernel),
                            hipFuncAttributeMaxDynamicSharedMemorySize, (int)shmem);

  sim_blood1d_kernel<<<1, NTHREADS, shmem, stream>>>(
      R1_param, A_init, Q_init, beta, A0, input_data, strides,
      (float*)d_out, K, T);
}